// SAFL_53326313947674
// MI455X (gfx1250) — compile-verified
//
#include <hip/hip_runtime.h>
#include <math.h>

// Problem constants (fixed by the reference setup)
#define B_    256
#define C_    2048
#define HW_   162          // 18*9
#define P_    6
#define NPP_  27           // patches per part
#define NTILES 11          // ceil(162/16)
#define NTHREADS 352       // 11 wave32s: one hw-tile per wave

typedef __attribute__((ext_vector_type(2))) float v2f;
typedef __attribute__((ext_vector_type(8))) float v8f;

// One workgroup per batch b.
//  Phase 0: stage part_tokens into LDS as K-pairs: tokp[k/2][p] = {T[k][p], T[k+1][p]}.
//  Phase 1: logits via V_WMMA_F32_16X16X4_F32:
//           D[hw, p] = sum_c X^T[hw, c] * T^T[c, p]; keep block-diagonal entries.
//           B columns p>=6 carry duplicated p=5 data (clamped index) — garbage
//           stays confined to unread D columns, so no predication needed.
//  Phase 1b: sigmoid + sum-normalize per part; emit attn of last part.
//  Phase 2: out[b,p,c] = sum_n attn[p,n] * x[b,c,p*27+n]  (second sweep of x[b],
//           hot in L2 because the same workgroup just read it).
__global__ __launch_bounds__(NTHREADS) void SAFL_part_attn_kernel(
    const float* __restrict__ x,          // [B, C, HW]
    const float* __restrict__ tokens,     // [P, C]
    float* __restrict__ out,              // [B, P, C]
    float* __restrict__ attn_last)        // [B, NPP]
{
    __shared__ float2 tokp[(C_ / 2) * P_];    // 48 KB: [k/2][p] -> {T[k][p], T[k+1][p]}
    __shared__ float  attn_lds[NTILES * 16];  // 176 floats (logits -> attn)
    __shared__ float  psum[P_];

    const int b   = blockIdx.x;
    const int tid = threadIdx.x;
    const float* __restrict__ xb = x + (size_t)b * C_ * HW_;

    // ---- Phase 0: tokens -> LDS as K-pairs ------------------------------
    for (int i = tid; i < (C_ / 2) * P_; i += NTHREADS) {
        const int p = i / (C_ / 2);
        const int j = i - p * (C_ / 2);                 // k-pair index
        const float2 pr = *(const float2*)(tokens + p * C_ + 2 * j); // 8B aligned
        tokp[j * P_ + p] = pr;
    }
    __syncthreads();

    // ---- Phase 1: WMMA logits ------------------------------------------
    const int lane = tid & 31;
    const int wave = tid >> 5;                // == tile index (11 waves, 11 tiles)
    const int half = lane >> 4;               // 0: K=0,1  1: K=2,3
    const int lr   = lane & 15;               // A: M index; B: N (=p) index
    const int lrp  = (lr < P_) ? lr : (P_ - 1);  // clamped p for B loads

    const int hw0 = wave * 16;
    int hwA = hw0 + lr;                       // A-matrix row (hw)
    if (hwA > HW_ - 1) hwA = HW_ - 1;         // clamp tail tile (cols unused)
    const int kbase = half * 2;

    v8f acc = {};
    #pragma unroll 4
    for (int c = 0; c < C_; c += 4) {
        // A tile: X^T, 16 hw x 4 c  (coalesced: 16 contiguous floats/half-wave)
        const float a0 = xb[(size_t)(c + kbase + 0) * HW_ + hwA];
        const float a1 = xb[(size_t)(c + kbase + 1) * HW_ + hwA];
        // B tile: T^T, 4 c x 16 p — one ds_load_b64 gives both K values
        const float2 bp = tokp[((c + kbase) >> 1) * P_ + lrp];
        v2f Av = {a0, a1};
        v2f Bv = {bp.x, bp.y};
        acc = __builtin_amdgcn_wmma_f32_16x16x4_f32(
            false, Av, false, Bv, (short)0, acc, false, false);
    }

    // D layout: VGPR r, lanes 0-15 -> M=r, N=lane; lanes 16-31 -> M=8+r.
    // logit[hw] lives at column p = hw/27: exactly one lane owns each hw.
    #pragma unroll
    for (int r = 0; r < 8; ++r) {
        const int hwr = hw0 + half * 8 + r;
        if (hwr < HW_ && lr == hwr / NPP_) attn_lds[hwr] = acc[r];
    }
    __syncthreads();

    // ---- Phase 1b: sigmoid + sum-normalize ------------------------------
    if (tid < HW_) attn_lds[tid] = 1.0f / (1.0f + expf(-attn_lds[tid]));
    __syncthreads();
    if (tid < P_) {
        float s = 0.0f;
        #pragma unroll
        for (int n = 0; n < NPP_; ++n) s += attn_lds[tid * NPP_ + n];
        psum[tid] = s;
    }
    __syncthreads();
    if (tid < HW_) attn_lds[tid] = attn_lds[tid] / psum[tid / NPP_];
    __syncthreads();
    if (tid < NPP_) attn_last[b * NPP_ + tid] = attn_lds[(P_ - 1) * NPP_ + tid];

    // ---- Phase 2: attention-weighted channel pooling --------------------
    float* __restrict__ ob = out + (size_t)b * P_ * C_;
    for (int c = tid; c < C_; c += NTHREADS) {
        const float* __restrict__ row = xb + (size_t)c * HW_;   // 162 contiguous
        float acc2[P_];
        #pragma unroll
        for (int p = 0; p < P_; ++p) {
            float s = 0.0f;
            #pragma unroll
            for (int n = 0; n < NPP_; ++n)
                s = fmaf(attn_lds[p * NPP_ + n], row[p * NPP_ + n], s);
            acc2[p] = s;
        }
        #pragma unroll
        for (int p = 0; p < P_; ++p) ob[p * C_ + c] = acc2[p];  // coalesced
    }
}

extern "C" void kernel_launch(void* const* d_in, const int* in_sizes, int n_in,
                              void* d_out, int out_size, void* d_ws, size_t ws_size,
                              hipStream_t stream) {
    (void)in_sizes; (void)n_in; (void)d_ws; (void)ws_size; (void)out_size;
    const float* x      = (const float*)d_in[0];   // [256, 2048, 18, 9]
    const float* tokens = (const float*)d_in[1];   // [6, 2048]
    float* out       = (float*)d_out;                          // [256, 6, 2048]
    float* attn_last = out + (size_t)B_ * P_ * C_;             // [256, 1, 27]
    SAFL_part_attn_kernel<<<B_, NTHREADS, 0, stream>>>(x, tokens, out, attn_last);
}